// GPUCalibrationOptimizer_7352984010758
// MI455X (gfx1250) — compile-verified
//
#include <hip/hip_runtime.h>
#include <hip/hip_bf16.h>

typedef float v2f __attribute__((ext_vector_type(2)));
typedef float v8f __attribute__((ext_vector_type(8)));

#define PEN_BIG 1.0e9f

// ---------------------------------------------------------------------------
// Kernel A: quaternion -> R, P = K*R, t' = K*t, reg term. One thread.
// ---------------------------------------------------------------------------
__global__ void calib_setup_kernel(const float* __restrict__ q,
                                   const float* __restrict__ tr,
                                   const float* __restrict__ K,
                                   float* __restrict__ pre) {
  if (threadIdx.x != 0 || blockIdx.x != 0) return;
  float qx = q[0], qy = q[1], qz = q[2], qw = q[3];
  float nq = sqrtf(qx * qx + qy * qy + qz * qz + qw * qw);
  float x = qx / nq, y = qy / nq, z = qz / nq, w = qw / nq;
  float R[9];
  R[0] = 1.f - 2.f * (y * y + z * z);
  R[1] = 2.f * (x * y - z * w);
  R[2] = 2.f * (x * z + y * w);
  R[3] = 2.f * (x * y + z * w);
  R[4] = 1.f - 2.f * (x * x + z * z);
  R[5] = 2.f * (y * z - x * w);
  R[6] = 2.f * (x * z - y * w);
  R[7] = 2.f * (y * z + x * w);
  R[8] = 1.f - 2.f * (x * x + y * y);
  for (int i = 0; i < 3; ++i)
    for (int j = 0; j < 3; ++j)
      pre[i * 3 + j] = K[i * 3 + 0] * R[0 * 3 + j] + K[i * 3 + 1] * R[1 * 3 + j] +
                       K[i * 3 + 2] * R[2 * 3 + j];
  for (int i = 0; i < 3; ++i)
    pre[9 + i] = K[i * 3 + 0] * tr[0] + K[i * 3 + 1] * tr[1] + K[i * 3 + 2] * tr[2];
  float dn = 1.f - nq;
  pre[12] = 0.01f * (tr[0] * tr[0] + tr[1] * tr[1] + tr[2] * tr[2] + dn * dn);
}

// ---------------------------------------------------------------------------
// Kernel B: per-frame pairwise masked-min distance, fully on the WMMA pipe.
//   Penalty WMMA:  A2 row m = [BIG, BIG*L[m], BIG*R[m], 0]
//                  B2 col n = [1, -vypos, -vyneg, 0]
//                  Pen[m][n] = BIG*(1 - L*vypos - R*vyneg)  in {0, BIG}
//   Distance WMMA: A  row m = [cx, cy, cx^2+cy^2, 1]
//                  B  col n = [-2ux, -2uy, 1, ux^2+uy^2]
//                  D = A*B + Pen = ||u-c||^2 (+BIG if invalid)
// Block = 256 threads (8 waves) = one frame; wave handles 16 points / iter.
// ---------------------------------------------------------------------------
__global__ __launch_bounds__(256) void calib_frame_kernel(
    const float* __restrict__ points,   // [F,4096,5]
    const float* __restrict__ bboxes,   // [F,128,4]
    const int* __restrict__ bleft,      // [F,128]
    const int* __restrict__ bright,     // [F,128]
    const float* __restrict__ pre,      // P[9], t'[3], reg
    float* __restrict__ frameOut)       // [64] loss, [64] has-flag
{
  const int f = blockIdx.x;
  const int tid = threadIdx.x;
  const int lane = tid & 31;
  const int wave = tid >> 5;
  const int upper = (lane >> 4) & 1;   // lanes 16-31 hold K=2,3
  const int lm = lane & 15;

  __shared__ float sA0[256];           // tile t: [0..15]=cx,      [16..31]=cnorm
  __shared__ float sA1[256];           // tile t: [0..15]=cy,      [16..31]=1.0
  __shared__ float sP0[256];           // tile t: [0..15]=BIG,     [16..31]=BIG*R
  __shared__ float sP1[256];           // tile t: [0..15]=BIG*L,   [16..31]=0
  __shared__ unsigned sAny;            // bit0 anyLeft, bit1 anyRight
  __shared__ float sRed[256];
  __shared__ float sCnt[256];

  if (tid == 0) sAny = 0u;
  __syncthreads();

  if (tid < 128) {
    const int m = tid;
    const float* bb = bboxes + ((size_t)f * 128 + m) * 4;
    float cx = bb[0], cy = bb[1];
    float cn = cx * cx + cy * cy;
    int t = m >> 4, r = m & 15;
    int L  = bleft[f * 128 + m] > 0;
    int Rv = bright[f * 128 + m] > 0;
    sA0[t * 32 + r]      = cx;
    sA0[t * 32 + 16 + r] = cn;
    sA1[t * 32 + r]      = cy;
    sA1[t * 32 + 16 + r] = 1.0f;
    sP0[t * 32 + r]      = PEN_BIG;
    sP0[t * 32 + 16 + r] = Rv ? PEN_BIG : 0.0f;
    sP1[t * 32 + r]      = L ? PEN_BIG : 0.0f;
    sP1[t * 32 + 16 + r] = 0.0f;
    unsigned ab = (unsigned)L | ((unsigned)Rv << 1);
    if (ab) atomicOr(&sAny, ab);
  }
  __syncthreads();

  // Projection (wave-uniform scalar loads).
  const float P0 = pre[0], P1 = pre[1], P2 = pre[2];
  const float P3 = pre[3], P4 = pre[4], P5 = pre[5];
  const float P6 = pre[6], P7 = pre[7], P8 = pre[8];
  const float T0 = pre[9], T1 = pre[10], T2 = pre[11];

  // Hoist frame-constant A matrices into registers.
  v2f A[8], A2[8];
#pragma unroll
  for (int t = 0; t < 8; ++t) {
    A[t].x  = sA0[t * 32 + lane];
    A[t].y  = sA1[t * 32 + lane];
    A2[t].x = sP0[t * 32 + lane];
    A2[t].y = sP1[t * 32 + lane];
  }
  const bool anyL = (sAny & 1u) != 0u;
  const bool anyR = (sAny & 2u) != 0u;

  float accS = 0.0f;
  float accC = 0.0f;

  const float* prow = points + (size_t)f * 4096 * 5;

  for (int it = 0; it < 32; ++it) {
    const int n = it * 128 + wave * 16 + lm;   // both halves mirror point lm
    const float* pp = prow + (size_t)n * 5;
    if (it < 31) __builtin_prefetch(pp + 128 * 5, 0, 1);
    float x = pp[0], y = pp[1], z = pp[2], vy = pp[4];

    float pc0 = P0 * x + P1 * y + P2 * z + T0;
    float pc1 = P3 * x + P4 * y + P5 * z + T1;
    float pc2 = P6 * x + P7 * y + P8 * z + T2;
    float inv = 1.0f / pc2;
    float ux = pc0 * inv, uy = pc1 * inv;
    float un = ux * ux + uy * uy;

    float vypos = (vy > 0.0f) ? 1.0f : 0.0f;
    float vyneg = (vy < 0.0f) ? 1.0f : 0.0f;

    v2f B;                               // distance B-matrix column
    B.x = upper ? 1.0f : -2.0f * ux;     // K=2 / K=0
    B.y = upper ? un   : -2.0f * uy;     // K=3 / K=1
    v2f B2;                              // penalty B-matrix column
    B2.x = upper ? -vyneg : 1.0f;        // K=2 / K=0
    B2.y = upper ? 0.0f   : -vypos;      // K=3 / K=1

    float dmin2 = 1e30f;
#pragma unroll
    for (int t = 0; t < 8; ++t) {
      v8f czero = {};
      // Pen in {0, BIG}: BIG - BIG cancels exactly, 0 stays exact.
      v8f pen = __builtin_amdgcn_wmma_f32_16x16x4_f32(
          false, A2[t], false, B2, (short)0, czero, false, false);
      // d^2 accumulated on top of Pen: valid pairs see C=0 -> exact d^2.
      v8f d = __builtin_amdgcn_wmma_f32_16x16x4_f32(
          false, A[t], false, B, (short)0, pen, false, false);
      float m0 = fminf(fminf(d[0], d[1]), fminf(d[2], d[3]));
      float m1 = fminf(fminf(d[4], d[5]), fminf(d[6], d[7]));
      dmin2 = fminf(dmin2, fminf(m0, m1));
    }
    // merge the two row-halves: lane n <-> lane n+16
    float other = __shfl_xor(dmin2, 16, 32);
    dmin2 = fminf(dmin2, other);

    bool pv = (vy > 0.0f) ? anyL : ((vy < 0.0f) ? anyR : false);
    float ds = pv ? sqrtf(dmin2) : 0.0f;
    float hub = (ds <= 50.0f) ? (0.5f * ds * ds) : (50.0f * (ds - 25.0f));
    // branch-free (keeps EXEC all-ones around WMMAs): lower half counts once
    float take = (upper == 0 && pv) ? 1.0f : 0.0f;
    accS += take * hub;
    accC += take;
  }

  // Deterministic block tree-reduction.
  sRed[tid] = accS;
  sCnt[tid] = accC;
  __syncthreads();
#pragma unroll
  for (int s = 128; s > 0; s >>= 1) {
    if (tid < s) {
      sRed[tid] += sRed[tid + s];
      sCnt[tid] += sCnt[tid + s];
    }
    __syncthreads();
  }
  if (tid == 0) {
    float cnt = sCnt[0];
    frameOut[f]      = (cnt > 0.0f) ? (sRed[0] / fmaxf(cnt, 1.0f)) : 0.0f;
    frameOut[64 + f] = (cnt > 0.0f) ? 1.0f : 0.0f;
  }
}

// ---------------------------------------------------------------------------
// Kernel C: serial deterministic frame combine + reg term.
// ---------------------------------------------------------------------------
__global__ void calib_final_kernel(const float* __restrict__ pre,
                                   const float* __restrict__ frameOut,
                                   float* __restrict__ out) {
  if (threadIdx.x != 0 || blockIdx.x != 0) return;
  float s = 0.0f, nf = 0.0f;
  for (int f = 0; f < 64; ++f) {
    s  += frameOut[f];
    nf += frameOut[64 + f];
  }
  out[0] = s / fmaxf(nf, 1.0f) + pre[12];
}

extern "C" void kernel_launch(void* const* d_in, const int* in_sizes, int n_in,
                              void* d_out, int out_size, void* d_ws, size_t ws_size,
                              hipStream_t stream) {
  const float* points = (const float*)d_in[0];  // [64,4096,5]
  const float* bboxes = (const float*)d_in[1];  // [64,128,4]
  const int* bleft    = (const int*)d_in[2];    // [64,128]
  const int* bright   = (const int*)d_in[3];    // [64,128]
  const float* quat   = (const float*)d_in[4];  // [4]
  const float* trans  = (const float*)d_in[5];  // [3]
  const float* cam    = (const float*)d_in[6];  // [3,3]

  float* pre      = (float*)d_ws;       // 16 floats: P[9], t'[3], reg
  float* frameOut = pre + 16;           // 128 floats: loss[64], has[64]

  calib_setup_kernel<<<1, 32, 0, stream>>>(quat, trans, cam, pre);
  calib_frame_kernel<<<64, 256, 0, stream>>>(points, bboxes, bleft, bright,
                                             pre, frameOut);
  calib_final_kernel<<<1, 32, 0, stream>>>(pre, frameOut, (float*)d_out);
}